// LinearSelfAttention_12008728560071
// MI455X (gfx1250) — compile-verified
//
#include <hip/hip_runtime.h>
#include <hip/hip_bf16.h>

// LinearSelfAttention (MobileViT-style) fused for MI455X / gfx1250.
// B=4, C=512, P=16, N=1024, E=1024, O=1+2E=2049. All fp32.
//
// Memory-bound (~206 GFLOP vs ~400MB minimal HBM traffic at 23.3TB/s), so we
// use exact fp32 WMMA (V_WMMA_F32_16X16X4_F32) and fuse:
//   cv = Wk @ (X @ softmax(q)) + bk      (softmax sums to 1 -> key GEMM eliminated)
//   out = Wout @ (relu(Wv@X + bv) * cv) + bout   (value tile lives in LDS only)
// gfx1250 specifics: wave32, WMMA f32 16x16x4, GLOBAL_LOAD_ASYNC_TO_LDS_B128
// for x-tile staging (ASYNCcnt), pair-interleaved u-tile for ds_*_b64 traffic.

#define EMBED 512
#define EXPAND 1024
#define NB 4
#define NP 16
#define NN 1024
#define BP (NB * NP)                 // 64 (b,p) pairs
#define X_BSTRIDE (EMBED * NP * NN)  // 8388608
#define X_CSTRIDE (NP * NN)          // 16384

typedef __attribute__((ext_vector_type(2))) float v2f;
typedef __attribute__((ext_vector_type(8))) float v8f;

// ---------------------------------------------------------------------------
// Kernel 1: per (b,p): query dot, softmax over N, weighted x-sum xs[c].
// One workgroup per (b,p). xs: [BP][EMBED] floats in workspace.
// ---------------------------------------------------------------------------
__global__ __launch_bounds__(256) void k1_query_softmax_xs(
    const float* __restrict__ x, const float* __restrict__ Wqkv,
    const float* __restrict__ bqkv, float* __restrict__ xs) {
  const int bp = blockIdx.x;              // 0..63
  const int b = bp >> 4, p = bp & 15;
  const float* xb = x + (size_t)b * X_BSTRIDE + (size_t)p * NN;
  __shared__ float s_q[NN];               // 4KB: query -> scores
  __shared__ float s_red[256];
  const int t = threadIdx.x;

  // query[n] = bq + sum_c Wq[c] * x[b,c,p,n]   (Wq = row 0 of Wqkv, scalarized)
  const float bq = bqkv[0];
  float q[4];
#pragma unroll
  for (int i = 0; i < 4; ++i) {
    const int n = t + i * 256;
    float acc = bq;
    for (int c = 0; c < EMBED; ++c)
      acc = fmaf(Wqkv[c], xb[(size_t)c * X_CSTRIDE + n], acc);
    q[i] = acc;
  }

  // softmax over the 1024 n-values (workgroup reduction)
  float m = fmaxf(fmaxf(q[0], q[1]), fmaxf(q[2], q[3]));
  s_red[t] = m;
  __syncthreads();
  for (int s = 128; s > 0; s >>= 1) {
    if (t < s) s_red[t] = fmaxf(s_red[t], s_red[t + s]);
    __syncthreads();
  }
  const float mx = s_red[0];
  __syncthreads();

  float e[4], psum = 0.f;
#pragma unroll
  for (int i = 0; i < 4; ++i) { e[i] = __expf(q[i] - mx); psum += e[i]; }
  s_red[t] = psum;
  __syncthreads();
  for (int s = 128; s > 0; s >>= 1) {
    if (t < s) s_red[t] += s_red[t + s];
    __syncthreads();
  }
  const float inv = 1.0f / s_red[0];
  __syncthreads();
#pragma unroll
  for (int i = 0; i < 4; ++i) s_q[t + i * 256] = e[i] * inv;
  __syncthreads();

  // xs[c] = sum_n score[n] * x[b,c,p,n]   (x re-read hits L2: x = 134MB < 192MB)
  for (int c = t; c < EMBED; c += 256) {
    const float* xr = xb + (size_t)c * X_CSTRIDE;
    float acc = 0.f;
    for (int n = 0; n < NN; ++n) acc = fmaf(s_q[n], xr[n], acc);
    xs[(size_t)bp * EMBED + c] = acc;
  }
}

// ---------------------------------------------------------------------------
// Kernel 2: cv[tok][e] = Wk @ xs + bk.  [1024x512] @ [512x64] via WMMA f32.
// 256 wave-tasks = 64 e-tiles x 4 token-tiles; 32 blocks of 8 waves.
// ---------------------------------------------------------------------------
__global__ __launch_bounds__(256) void k2_context_vector(
    const float* __restrict__ Wqkv, const float* __restrict__ bqkv,
    const float* __restrict__ xs, float* __restrict__ cv) {
  const int lane = threadIdx.x & 31;
  const int wid  = (blockIdx.x * 256 + threadIdx.x) >> 5;  // 0..255
  const int et = wid >> 2;      // e-tile 0..63
  const int tt = wid & 3;       // token-tile 0..3
  const int mrow = lane & 15;
  const int hi   = lane >> 4;   // 0/1 -> K pair {0,1} or {2,3}
  const int koff = hi * 2;

  const float* Wk  = Wqkv + (size_t)(1 + et * 16 + mrow) * EMBED;  // A row (M)
  const float* xsn = xs + (size_t)(tt * 16 + mrow) * EMBED;        // B col (N)
  v8f acc = {0.f, 0.f, 0.f, 0.f, 0.f, 0.f, 0.f, 0.f};
  for (int kb = 0; kb < EMBED; kb += 4) {
    v2f a = {Wk[kb + koff], Wk[kb + koff + 1]};
    v2f bb = {xsn[kb + koff], xsn[kb + koff + 1]};
    acc = __builtin_amdgcn_wmma_f32_16x16x4_f32(false, a, false, bb,
                                                (short)0, acc, false, false);
  }
#pragma unroll
  for (int r = 0; r < 8; ++r) {
    const int e = et * 16 + r + hi * 8;       // C/D layout: VGPR r -> M=r (+8 hi-half)
    const int tok = tt * 16 + mrow;
    cv[(size_t)tok * EXPAND + e] = acc[r] + bqkv[1 + e];
  }
}

// ---------------------------------------------------------------------------
// Kernel 3: per 16-token tile: u = relu(Wv@xtile + bv)*cv in LDS, then
// out_tile = Wout@u + bout.  8 waves; stage1: 128 e-rows/wave, stage2: 64 o-rows/wave.
// Dynamic LDS: x tile 32KB (natural [c][n]) + u tile 64KB (pair-interleaved
// [e/2][n][2] so stage-2 B operands are single ds_load_b64). 96KB < 320KB/WGP.
// x tile staged with GLOBAL_LOAD_ASYNC_TO_LDS_B128 (no VGPR roundtrip).
// ---------------------------------------------------------------------------
__global__ __launch_bounds__(256) void k3_value_out(
    const float* __restrict__ x, const float* __restrict__ Wqkv,
    const float* __restrict__ bqkv, const float* __restrict__ Wout,
    const float* __restrict__ bout, const float* __restrict__ cv,
    float* __restrict__ out) {
  extern __shared__ float smem[];
  float* s_x = smem;               // [512][16]          natural layout
  float* s_u = smem + EMBED * 16;  // [512][16][2]       pair-interleaved over e

  const int g = blockIdx.x;        // 0..4095
  const int bp = g >> 6;
  const int n0 = (g & 63) * 16;
  const int b = bp >> 4, p = bp & 15;
  const size_t xbase = (size_t)b * X_BSTRIDE + (size_t)p * NN + n0;

  const int t = threadIdx.x;

  // ---- async stage of the 512x16 x tile: 8 rounds x 256 lanes x 16B --------
  // Low 32 bits of a generic LDS pointer are the LDS byte offset on AMDGPU.
  const unsigned lds_x_base = (unsigned)(uintptr_t)s_x;
#pragma unroll
  for (int r = 0; r < 8; ++r) {
    const int f = (t + r * 256) * 4;           // float index within tile
    const int c = f >> 4, n = f & 15;          // n in {0,4,8,12}: 16B aligned
    const float* gp = x + xbase + (size_t)c * X_CSTRIDE + n;
    const unsigned ldsb = lds_x_base + (unsigned)(f * 4);
    asm volatile("global_load_async_to_lds_b128 %0, %1, off"
                 :: "v"(ldsb), "v"(gp) : "memory");
  }
  asm volatile("s_wait_asynccnt 0x0" ::: "memory");
  __syncthreads();

  const int lane = t & 31;         // wave32
  const int wave = t >> 5;         // 0..7
  const int mrow = lane & 15;
  const int hi   = lane >> 4;
  const int koff = hi * 2;
  const float* cvp = cv + (size_t)bp * EXPAND;

  // ---- Stage 1: value GEMM + relu + cv scale -> s_u (paired layout) --------
  {
    const int e0 = wave * 128;
    const float* WvBase = Wqkv + (size_t)(1 + EXPAND) * EMBED;  // value rows
    v8f acc[8];
#pragma unroll
    for (int mt = 0; mt < 8; ++mt) acc[mt] = (v8f){0.f,0.f,0.f,0.f,0.f,0.f,0.f,0.f};
    for (int kb = 0; kb < EMBED; kb += 4) {
      v2f bb = {s_x[(kb + koff) * 16 + mrow], s_x[(kb + koff + 1) * 16 + mrow]};
#pragma unroll
      for (int mt = 0; mt < 8; ++mt) {
        const float* ar = WvBase + (size_t)(e0 + mt * 16 + mrow) * EMBED + kb + koff;
        v2f a = {ar[0], ar[1]};     // 8B contiguous load, L2-resident weight
        acc[mt] = __builtin_amdgcn_wmma_f32_16x16x4_f32(false, a, false, bb,
                                                        (short)0, acc[mt], false, false);
      }
    }
#pragma unroll
    for (int mt = 0; mt < 8; ++mt) {
#pragma unroll
      for (int r = 0; r < 8; r += 2) {         // e even: write (e, e+1) as b64
        const int e = e0 + mt * 16 + r + hi * 8;
        const float v0 = fmaxf(acc[mt][r]     + bqkv[1 + EXPAND + e],     0.f) * cvp[e];
        const float v1 = fmaxf(acc[mt][r + 1] + bqkv[1 + EXPAND + e + 1], 0.f) * cvp[e + 1];
        *(v2f*)(s_u + (e >> 1) * 32 + mrow * 2) = (v2f){v0, v1};
      }
    }
  }
  __syncthreads();

  // ---- Stage 2: out GEMM: Wout[512x1024] @ u[1024x16] + bout ----------------
  {
    const int o0 = wave * 64;
    v8f acc[4];
#pragma unroll
    for (int mt = 0; mt < 4; ++mt) acc[mt] = (v8f){0.f,0.f,0.f,0.f,0.f,0.f,0.f,0.f};
    for (int kb = 0; kb < EXPAND; kb += 4) {
      const int k = kb + koff;                 // even
      const v2f bb = *(const v2f*)(s_u + (k >> 1) * 32 + mrow * 2);  // ds_load_b64
#pragma unroll
      for (int mt = 0; mt < 4; ++mt) {
        const float* ar = Wout + (size_t)(o0 + mt * 16 + mrow) * EXPAND + k;
        v2f a = {ar[0], ar[1]};
        acc[mt] = __builtin_amdgcn_wmma_f32_16x16x4_f32(false, a, false, bb,
                                                        (short)0, acc[mt], false, false);
      }
    }
#pragma unroll
    for (int mt = 0; mt < 4; ++mt) {
#pragma unroll
      for (int r = 0; r < 8; ++r) {
        const int o = o0 + mt * 16 + r + hi * 8;
        out[(size_t)b * X_BSTRIDE + (size_t)o * X_CSTRIDE + (size_t)p * NN + n0 + mrow] =
            acc[mt][r] + bout[o];
      }
    }
  }
}

// ---------------------------------------------------------------------------
extern "C" void kernel_launch(void* const* d_in, const int* in_sizes, int n_in,
                              void* d_out, int out_size, void* d_ws, size_t ws_size,
                              hipStream_t stream) {
  const float* x    = (const float*)d_in[0];
  const float* Wqkv = (const float*)d_in[1];
  const float* bqkv = (const float*)d_in[2];
  const float* Wout = (const float*)d_in[3];
  const float* bout = (const float*)d_in[4];
  float* out = (float*)d_out;

  float* xs = (float*)d_ws;                  // [64][512]  = 128KB
  float* cv = xs + (size_t)BP * EMBED;       // [64][1024] = 256KB

  k1_query_softmax_xs<<<BP, 256, 0, stream>>>(x, Wqkv, bqkv, xs);
  k2_context_vector<<<32, 256, 0, stream>>>(Wqkv, bqkv, xs, cv);
  const size_t lds3 = (size_t)(EMBED * 16 + EXPAND * 16) * sizeof(float);  // 96KB
  k3_value_out<<<BP * (NN / 16), 256, lds3, stream>>>(x, Wqkv, bqkv, Wout, bout, cv, out);
}